// BatchedCGSolver_82789789598016
// MI455X (gfx1250) — compile-verified
//
#include <hip/hip_runtime.h>
#include <cstdint>
#include <cstddef>

// ---------------- problem constants (match reference) ----------------
#define NVV      50000
#define NEE      200000
#define BB       8
#define V3C      (NVV * 3)          // 150000 elements per batch vector
#define NTC      ((size_t)BB * V3C) // 1.2M elements per full state vector
#define CG_ITERS 25
#define EPSF     1e-8f
#define H2K      0.1f               // H*H*K_STIFF = 1e-4 * 1000
#define TPB      256
#define NB       128                // reduction blocks per batch (fixed)
#define ETILE    256                // edges per LDS tile
#define EBLOCKS  208

// ---------------- CDNA5 async global->LDS copy (guarded) ----------------
#if __has_builtin(__builtin_amdgcn_global_load_async_to_lds_b128)
#define USE_ASYNC_LDS 1
#endif

typedef int   v4i_vs __attribute__((vector_size(16)));
typedef float v2f __attribute__((ext_vector_type(2)));
typedef float v8f __attribute__((ext_vector_type(8)));

__device__ __forceinline__ void async_copy16(void* lds_dst, const void* g_src) {
#if defined(USE_ASYNC_LDS)
  __builtin_amdgcn_global_load_async_to_lds_b128(
      (__attribute__((address_space(1))) v4i_vs*)(g_src),
      (__attribute__((address_space(3))) v4i_vs*)(lds_dst),
      0, 0);
#else
  *(float4*)lds_dst = *(const float4*)g_src;   // synchronous fallback
#endif
}

__device__ __forceinline__ void wait_async_le2() {
#if defined(USE_ASYNC_LDS)
#if __has_builtin(__builtin_amdgcn_s_wait_asynccnt)
  __builtin_amdgcn_s_wait_asynccnt(2);
#else
  asm volatile("s_wait_asynccnt 0x2" ::: "memory");
#endif
#endif
}

__device__ __forceinline__ void wait_async_le0() {
#if defined(USE_ASYNC_LDS)
#if __has_builtin(__builtin_amdgcn_s_wait_asynccnt)
  __builtin_amdgcn_s_wait_asynccnt(0);
#else
  asm volatile("s_wait_asynccnt 0x0" ::: "memory");
#endif
#endif
}

// fire-and-forget fp32 global atomic add -> global_atomic_add_f32 (no return)
__device__ __forceinline__ void atomicAddRelaxed(float* addr, float v) {
  __hip_atomic_fetch_add(addr, v, __ATOMIC_RELAXED, __HIP_MEMORY_SCOPE_AGENT);
}

// deterministic intra-block reduction (wave32 shuffles + fixed-order LDS sum)
__device__ __forceinline__ float blockReduceSum256(float v, float* sm) {
#pragma unroll
  for (int o = 16; o > 0; o >>= 1) v += __shfl_down(v, o, 32);
  const int lane = threadIdx.x & 31;
  const int w    = threadIdx.x >> 5;
  if (lane == 0) sm[w] = v;
  __syncthreads();
  float s = 0.0f;
  if (threadIdx.x == 0) {
#pragma unroll
    for (int k = 0; k < 8; ++k) s += sm[k];
  }
  return s;   // valid in thread 0 only
}

// ---------------- kernels ----------------

// Precompute per-edge record: {nx,ny,nz,kd,kn,bits(i),bits(j),0}
// J*dv = kd*dv + kn*(n.dv)*n with kd=H^2*K*(1-ratio), kn=H^2*K*ratio
__global__ void edge_setup_k(const float* __restrict__ pos,
                             const int*   __restrict__ edges,
                             const float* __restrict__ rl,
                             float* __restrict__ erec) {
  int e = blockIdx.x * blockDim.x + threadIdx.x;
  if (e >= NEE) return;
  int i = edges[2 * e + 0];
  int j = edges[2 * e + 1];
  float dx = pos[3 * j + 0] - pos[3 * i + 0];
  float dy = pos[3 * j + 1] - pos[3 * i + 1];
  float dz = pos[3 * j + 2] - pos[3 * i + 2];
  float l   = sqrtf(dx * dx + dy * dy + dz * dz);
  float inv = 1.0f / (l + EPSF);
  float ratio = rl[e] * inv;
  float* o = erec + (size_t)e * 8;
  o[0] = dx * inv;
  o[1] = dy * inv;
  o[2] = dz * inv;
  o[3] = H2K * (1.0f - ratio);
  o[4] = H2K * ratio;
  o[5] = __int_as_float(i);
  o[6] = __int_as_float(j);
  o[7] = 0.0f;
}

// x=0, r=p=rhs, Ap=mass*rhs, partial sum of rhs^2 per (batch, block)
__global__ void cg_init_k(const float* __restrict__ rhs,
                          const float* __restrict__ mass,
                          float* __restrict__ x,  float* __restrict__ r,
                          float* __restrict__ p,  float* __restrict__ Ap,
                          float* __restrict__ part) {
  const int bat = blockIdx.y;
  const size_t base = (size_t)bat * V3C;
  float acc = 0.0f;
  for (int idx = blockIdx.x * blockDim.x + threadIdx.x; idx < V3C;
       idx += gridDim.x * blockDim.x) {
    float v = rhs[base + idx];
    x[base + idx]  = 0.0f;
    r[base + idx]  = v;
    p[base + idx]  = v;
    Ap[base + idx] = mass[idx / 3] * v;
    acc = fmaf(v, v, acc);
  }
  __shared__ float sm[8];
  float s = blockReduceSum256(acc, sm);
  if (threadIdx.x == 0) part[bat * NB + blockIdx.x] = s;
}

// Final reduction stage on the matrix pipe: dst[b] = sum_{k<128} part[b*128+k].
// One wave (EXEC all ones). 16 rows = (batch b = m>>1, half h = m&1), each row
// holds 64 partials; 16 chained V_WMMA_F32_16X16X4_F32 with all-ones B compute
// all 16 row sums at once. A-layout per ISA: lanes 0-15 = M rows, VGPR0/1 =
// K0/K1; lanes 16-31 = K2/K3. D-layout: VGPR r -> M=r (lanes 0-15), M=r+8
// (lanes 16-31).
__global__ void finish_reduce_wmma_k(const float* __restrict__ part,
                                     float* __restrict__ dst) {
#if defined(__HIP_DEVICE_COMPILE__)
  const int L = threadIdx.x;          // 0..31, one wave, no divergence
  const int m = L & 15;               // logical row
  const int b = m >> 1;               // batch
  const int h = m & 1;                // which half of the 128 partials
  const int cofs = (L >= 16) ? 2 : 0; // K2/K3 live in lanes 16-31
  const float* rowp = part + b * NB + h * 64;
  v2f ones; ones.x = 1.0f; ones.y = 1.0f;
  v8f c = {};
#pragma unroll
  for (int t = 0; t < 16; ++t) {      // 16 * K4 = 64 columns per row
    v2f a;
    a.x = rowp[4 * t + cofs + 0];
    a.y = rowp[4 * t + cofs + 1];
    c = __builtin_amdgcn_wmma_f32_16x16x4_f32(false, a, false, ones,
                                              (short)0, c, false, false);
  }
  if (L == 0) {                       // rows 0..7 -> batches 0..3
#pragma unroll
    for (int bb = 0; bb < 4; ++bb) dst[bb] = c[2 * bb] + c[2 * bb + 1];
  } else if (L == 16) {               // rows 8..15 -> batches 4..7
#pragma unroll
    for (int bb = 4; bb < 8; ++bb) dst[bb] = c[2 * bb - 8] + c[2 * bb - 7];
  }
#endif
}

// Sparse matvec scatter: Ap[:,i] += c, Ap[:,j] -= c, c = kd*dv + kn*(n.dv)*n
// Edge records are streamed global->LDS with ASYNCcnt double buffering.
__global__ void edge_apply_k(const float* __restrict__ erec,
                             const float* __restrict__ p,
                             float* __restrict__ Ap) {
  __shared__ float lbuf[2][ETILE * 8];   // 16 KB
  const int tid    = threadIdx.x;
  const int ntiles = (NEE + ETILE - 1) / ETILE;
  int tile = blockIdx.x;
  if (tile >= ntiles) return;

  { // prologue: fetch first tile into buffer 0
    int ec = tile * ETILE + tid; if (ec > NEE - 1) ec = NEE - 1;
    async_copy16(&lbuf[0][tid * 8 + 0], erec + (size_t)ec * 8 + 0);
    async_copy16(&lbuf[0][tid * 8 + 4], erec + (size_t)ec * 8 + 4);
  }
  int parity = 0;
  for (; tile < ntiles; tile += gridDim.x) {
    const int nxt = tile + gridDim.x;
    if (nxt < ntiles) { // prefetch next tile into the other buffer
      int ec = nxt * ETILE + tid; if (ec > NEE - 1) ec = NEE - 1;
      async_copy16(&lbuf[parity ^ 1][tid * 8 + 0], erec + (size_t)ec * 8 + 0);
      async_copy16(&lbuf[parity ^ 1][tid * 8 + 4], erec + (size_t)ec * 8 + 4);
      wait_async_le2();   // current tile's 2 copies complete; next tile's may fly
    } else {
      wait_async_le0();
    }

    const int e = tile * ETILE + tid;
    if (e < NEE) {
      const float* rc = &lbuf[parity][tid * 8];
      const float nx = rc[0], ny = rc[1], nz = rc[2];
      const float kd = rc[3], kn = rc[4];
      const int vi = __float_as_int(rc[5]);
      const int vj = __float_as_int(rc[6]);
      const size_t oi = (size_t)vi * 3;
      const size_t oj = (size_t)vj * 3;
#pragma unroll
      for (int b = 0; b < BB; ++b) {
        const size_t base = (size_t)b * V3C;
        const float* pi = p + base + oi;
        const float* pj = p + base + oj;
        float dx = pi[0] - pj[0];
        float dy = pi[1] - pj[1];
        float dz = pi[2] - pj[2];
        float s  = kn * fmaf(nx, dx, fmaf(ny, dy, nz * dz));
        float cx = fmaf(kd, dx, s * nx);
        float cy = fmaf(kd, dy, s * ny);
        float cz = fmaf(kd, dz, s * nz);
        float* ai = Ap + base + oi;
        float* aj = Ap + base + oj;
        atomicAddRelaxed(ai + 0,  cx);
        atomicAddRelaxed(ai + 1,  cy);
        atomicAddRelaxed(ai + 2,  cz);
        atomicAddRelaxed(aj + 0, -cx);
        atomicAddRelaxed(aj + 1, -cy);
        atomicAddRelaxed(aj + 2, -cz);
      }
    }
    parity ^= 1;
  }
}

// Partial dot(p, Ap) per (batch, block), float4 vectorized
__global__ void dot_partial_k(const float* __restrict__ a,
                              const float* __restrict__ b,
                              float* __restrict__ part) {
  const int bat = blockIdx.y;
  const float4* a4 = (const float4*)(a + (size_t)bat * V3C);
  const float4* b4 = (const float4*)(b + (size_t)bat * V3C);
  const int n4 = V3C / 4;   // 37500
  float acc = 0.0f;
  for (int i = blockIdx.x * blockDim.x + threadIdx.x; i < n4;
       i += gridDim.x * blockDim.x) {
    float4 x = a4[i], y = b4[i];
    acc = fmaf(x.x, y.x, fmaf(x.y, y.y, fmaf(x.z, y.z, fmaf(x.w, y.w, acc))));
  }
  __shared__ float sm[8];
  float s = blockReduceSum256(acc, sm);
  if (threadIdx.x == 0) part[bat * NB + blockIdx.x] = s;
}

// alpha = rs/(pAp+eps); x += alpha p; r -= alpha Ap; partial sum r^2
__global__ void update_xr_k(float* __restrict__ x, float* __restrict__ r,
                            const float* __restrict__ p, const float* __restrict__ Ap,
                            const float* __restrict__ rsArr,
                            const float* __restrict__ pApArr,
                            float* __restrict__ part, int it) {
  const int bat = blockIdx.y;
  const float alpha = rsArr[it * BB + bat] / (pApArr[it * BB + bat] + EPSF);
  const size_t base = (size_t)bat * V3C;
  float4* x4 = (float4*)(x + base);
  float4* r4 = (float4*)(r + base);
  const float4* p4  = (const float4*)(p + base);
  const float4* a4  = (const float4*)(Ap + base);
  const int n4 = V3C / 4;
  float acc = 0.0f;
  for (int i = blockIdx.x * blockDim.x + threadIdx.x; i < n4;
       i += gridDim.x * blockDim.x) {
    float4 pp = p4[i], aa = a4[i], xx = x4[i], rr = r4[i];
    xx.x = fmaf(alpha, pp.x, xx.x); rr.x = fmaf(-alpha, aa.x, rr.x);
    xx.y = fmaf(alpha, pp.y, xx.y); rr.y = fmaf(-alpha, aa.y, rr.y);
    xx.z = fmaf(alpha, pp.z, xx.z); rr.z = fmaf(-alpha, aa.z, rr.z);
    xx.w = fmaf(alpha, pp.w, xx.w); rr.w = fmaf(-alpha, aa.w, rr.w);
    acc = fmaf(rr.x, rr.x, fmaf(rr.y, rr.y, fmaf(rr.z, rr.z, fmaf(rr.w, rr.w, acc))));
    x4[i] = xx; r4[i] = rr;
  }
  __shared__ float sm[8];
  float s = blockReduceSum256(acc, sm);
  if (threadIdx.x == 0) part[bat * NB + blockIdx.x] = s;
}

// beta = rs_new/(rs+eps); p = r + beta p; Ap = mass*p (prep next matvec)
__global__ void update_p_apinit_k(float* __restrict__ p, const float* __restrict__ r,
                                  float* __restrict__ Ap, const float* __restrict__ mass,
                                  const float* __restrict__ rsArr, int it) {
  const int bat = blockIdx.y;
  const float beta = rsArr[(it + 1) * BB + bat] / (rsArr[it * BB + bat] + EPSF);
  const size_t base = (size_t)bat * V3C;
  for (int idx = blockIdx.x * blockDim.x + threadIdx.x; idx < V3C;
       idx += gridDim.x * blockDim.x) {
    float pv = fmaf(beta, p[base + idx], r[base + idx]);
    p[base + idx]  = pv;
    Ap[base + idx] = mass[idx / 3] * pv;
  }
}

// ---------------- host launcher ----------------
extern "C" void kernel_launch(void* const* d_in, const int* in_sizes, int n_in,
                              void* d_out, int out_size, void* d_ws, size_t ws_size,
                              hipStream_t stream) {
  const float* rhs  = (const float*)d_in[0];
  const float* pos  = (const float*)d_in[1];
  const int*   edges= (const int*)  d_in[2];
  const float* rl   = (const float*)d_in[3];
  const float* mass = (const float*)d_in[4];
  float* x  = (float*)d_out;                 // (B, NV, 3) solution
  float* ws = (float*)d_ws;

  // workspace layout (floats): r | p | Ap | erec | pAp[it][B] | rs[it][B] | partials
  float* r      = ws;
  float* p      = ws + NTC;
  float* Ap     = ws + 2 * NTC;
  float* erec   = ws + 3 * NTC;                       // NE*8
  float* pApArr = erec + (size_t)NEE * 8;             // CG_ITERS*B
  float* rsArr  = pApArr + (size_t)CG_ITERS * BB;     // (CG_ITERS+1)*B
  float* part   = rsArr + (size_t)(CG_ITERS + 1) * BB;// NB*B

  const dim3 gvec(NB, BB);

  edge_setup_k<<<(NEE + TPB - 1) / TPB, TPB, 0, stream>>>(pos, edges, rl, erec);
  cg_init_k<<<gvec, TPB, 0, stream>>>(rhs, mass, x, r, p, Ap, part);
  finish_reduce_wmma_k<<<1, 32, 0, stream>>>(part, rsArr);   // rs[0]

  for (int it = 0; it < CG_ITERS; ++it) {
    edge_apply_k<<<EBLOCKS, ETILE, 0, stream>>>(erec, p, Ap);
    dot_partial_k<<<gvec, TPB, 0, stream>>>(p, Ap, part);
    finish_reduce_wmma_k<<<1, 32, 0, stream>>>(part, pApArr + (size_t)it * BB);
    update_xr_k<<<gvec, TPB, 0, stream>>>(x, r, p, Ap, rsArr, pApArr, part, it);
    finish_reduce_wmma_k<<<1, 32, 0, stream>>>(part, rsArr + (size_t)(it + 1) * BB);
    if (it + 1 < CG_ITERS)
      update_p_apinit_k<<<gvec, TPB, 0, stream>>>(p, r, Ap, mass, rsArr, it);
  }
}